// OptimizedS4Classifier_24352464569723
// MI455X (gfx1250) — compile-verified
//
#include <hip/hip_runtime.h>
#include <math.h>

typedef __attribute__((ext_vector_type(2))) float v2f;
typedef __attribute__((ext_vector_type(8))) float v8f;
typedef int v4i __attribute__((vector_size(16)));  // matches builtin prototype

// fp32 WMMA: D(16x16,f32) = A(16x4,f32) x B(4x16,f32) + C
#define WMMA_F32(a, b, c) \
  __builtin_amdgcn_wmma_f32_16x16x4_f32(false, (a), false, (b), (short)0, (c), false, false)

// Async memory->LDS path (gfx1250): guarded so compile never regresses.
#if defined(__HIP_DEVICE_COMPILE__) &&                                     \
    __has_builtin(__builtin_amdgcn_global_load_async_to_lds_b128) &&       \
    __has_builtin(__builtin_amdgcn_s_wait_asynccnt)
#define USE_ASYNC_LDS 1
#else
#define USE_ASYNC_LDS 0
#endif

#if USE_ASYNC_LDS
#define GLOBAL_AS __attribute__((address_space(1)))
#define LDS_AS    __attribute__((address_space(3)))
// memory -> LDS, 16 bytes per lane, tracked by ASYNCcnt
#define ASYNC_G2L_B128(gp, lp)                                             \
  __builtin_amdgcn_global_load_async_to_lds_b128(                          \
      (GLOBAL_AS v4i*)(gp), (LDS_AS v4i*)(lp), 0, 0)
#endif

static constexpr int   Bn  = 128;   // batch
static constexpr int   Cc  = 64;    // channels
static constexpr int   Ll  = 4000;  // raw length
static constexpr int   Tt  = 2000;  // downsampled length
static constexpr int   Dd  = 128;   // d_model
static constexpr int   Ss  = 16;    // state
static constexpr float DTC = 1e-4f;
static constexpr float EPS = 1e-5f;

// ---------------------------------------------------------------------------
// Kernel 1: fused Conv1d(k=3,s=2,p=1) + BN(eval) + ReLU + Linear(C->D) + pos.
// One block = (batch b, 64 consecutive t). Conv result stays in LDS; the
// projection is done with fp32 WMMA (K=64 -> 16 chained 16x16x4 steps).
// ---------------------------------------------------------------------------
__global__ __launch_bounds__(128)
void s4_conv_proj(const float* __restrict__ x, const float* __restrict__ convw,
                  const float* __restrict__ convb, const float* __restrict__ bng,
                  const float* __restrict__ bnb, const float* __restrict__ Win,
                  const float* __restrict__ bin, const float* __restrict__ pos,
                  float* __restrict__ hin) {
  __shared__ float xwin[64][132];  // [ci][129 used] input window
  __shared__ float xd[64][68];     // [t][ci] conv output tile

  const int blk = blockIdx.x;
  const int b   = blk >> 5;   // 32 t-chunks per batch
  const int tc  = blk & 31;
  const int t0  = tc * 64;
  const int tid = threadIdx.x;

  // load x window: l in [2*t0-1, 2*t0+127]
  for (int i = tid; i < 64 * 129; i += 128) {
    int ci = i / 129, j = i - ci * 129;
    int l  = 2 * t0 - 1 + j;
    float v = 0.f;
    if (l >= 0 && l < Ll) v = x[(size_t)b * Cc * Ll + (size_t)ci * Ll + l];
    xwin[ci][j] = v;
  }
  __syncthreads();

  const float bnscale = rsqrtf(1.0f + EPS);
  for (int idx = tid; idx < 64 * 64; idx += 128) {
    int t = idx >> 6, co = idx & 63;
    float acc = convb[co];
    const float* wrow = convw + co * Cc * 3;
    for (int ci = 0; ci < 64; ++ci) {
      acc += wrow[ci * 3 + 0] * xwin[ci][2 * t + 0];
      acc += wrow[ci * 3 + 1] * xwin[ci][2 * t + 1];
      acc += wrow[ci * 3 + 2] * xwin[ci][2 * t + 2];
    }
    float v = bng[co] * acc * bnscale + bnb[co];
    v = fmaxf(v, 0.f);
    if (t0 + t >= Tt) v = 0.f;  // tail guard (2000 not multiple of 64)
    xd[t][co] = v;
  }
  __syncthreads();

  // WMMA projection: wave wv owns t-rows [16wv,16wv+16), loops 8 N-tiles.
  const int wv = tid >> 5, lane = tid & 31;
  const int m = lane & 15, kh = lane >> 4;
  const bool full = (t0 + 64 <= Tt);  // block-uniform tail check
  for (int nt = 0; nt < 8; ++nt) {
    v8f c = {};
    for (int kk = 0; kk < 16; ++kk) {
      int kb = kk * 4 + kh * 2;
      v2f a, bb;
      a.x  = xd[wv * 16 + m][kb];
      a.y  = xd[wv * 16 + m][kb + 1];
      int dcol = nt * 16 + m;
      bb.x = Win[(size_t)kb * Dd + dcol];
      bb.y = Win[(size_t)(kb + 1) * Dd + dcol];
      c = WMMA_F32(a, bb, c);
    }
    if (full) {
#pragma unroll
      for (int i = 0; i < 8; ++i) {
        int row = wv * 16 + i + kh * 8;
        int col = nt * 16 + m;
        int gt  = t0 + row;
        hin[(size_t)b * Tt * Dd + (size_t)gt * Dd + col] =
            c[i] + bin[col] + pos[(size_t)gt * Dd + col];
      }
    } else {
#pragma unroll
      for (int i = 0; i < 8; ++i) {
        int row = wv * 16 + i + kh * 8;
        int col = nt * 16 + m;
        int gt  = t0 + row;
        if (gt < Tt)
          hin[(size_t)b * Tt * Dd + (size_t)gt * Dd + col] =
              c[i] + bin[col] + pos[(size_t)gt * Dd + col];
      }
    }
  }
}

// ---------------------------------------------------------------------------
// Kernel 2: sequential S4 scan, fused with both LayerNorms, residual+ReLU and
// the time-mean pool. One block = 16 batch rows, 8 waves (256 threads).
// Per step: async-load u tile to LDS -> P = U·B_discT (K split over waves,
// WMMA, LDS reduce) -> h update -> Y = H·CT (N split over waves, WMMA) ->
// LN1 -> relu(u+s4) -> LN2 -> pool accumulate.
// ---------------------------------------------------------------------------
__global__ __launch_bounds__(256)
void s4_scan(const float* __restrict__ hin, const float* __restrict__ Aarr,
             const float* __restrict__ Bmat, const float* __restrict__ Cmat,
             const float* __restrict__ Dvec, const float* __restrict__ g1,
             const float* __restrict__ be1, const float* __restrict__ g2,
             const float* __restrict__ be2, float* __restrict__ pooled) {
  __shared__ float utile[16][132];
  __shared__ float ytile[16][132];
  __shared__ float pred[8][16][16];
  __shared__ float htile[16][16];
  __shared__ float rstat[16][16][2];
  __shared__ float mstat[16][2];
  __shared__ float adisc[16];
  __shared__ float dv[128], sg1[128], sb1[128], sg2[128], sb2[128];

  const int tid  = threadIdx.x;
  const int wv   = tid >> 5, lane = tid & 31;
  const int m    = lane & 15, kh = lane >> 4;
  const int b0   = blockIdx.x * 16;
  const int row  = tid >> 4, seg = tid & 15;

  if (tid < 16) adisc[tid] = expf(-DTC * fabsf(Aarr[tid]));
  if (tid < 128) {
    dv[tid]  = Dvec[tid];
    sg1[tid] = g1[tid];  sb1[tid] = be1[tid];
    sg2[tid] = g2[tid];  sb2[tid] = be2[tid];
  }
  htile[row][seg] = 0.f;

  // Constant B-fragments: bf for B_discT[d][s]=DT*Bmat[s][d] (wave's K chunk),
  // cf for CT[s][d]=Cmat[d][s] (wave's N chunk, K=16).
  v2f bf[4], cf[4];
#pragma unroll
  for (int kk = 0; kk < 4; ++kk) {
    int kb = wv * 16 + kk * 4 + kh * 2;          // d row of B_discT
    bf[kk].x = DTC * Bmat[(size_t)m * Dd + kb];
    bf[kk].y = DTC * Bmat[(size_t)m * Dd + kb + 1];
    int kb2 = kk * 4 + kh * 2;                   // s row of CT
    int nd  = wv * 16 + m;                       // d column
    cf[kk].x = Cmat[(size_t)nd * Ss + kb2];
    cf[kk].y = Cmat[(size_t)nd * Ss + kb2 + 1];
  }
  __syncthreads();

  float acc[8] = {0, 0, 0, 0, 0, 0, 0, 0};

  for (int t = 0; t < Tt; ++t) {
    // 1. bring u_t tile into LDS
    const float* urow =
        hin + (size_t)(b0 + row) * Tt * Dd + (size_t)t * Dd + seg * 8;
#if USE_ASYNC_LDS
    {
      // two 16B async chunks per thread: memory -> LDS, no VGPR staging
      ASYNC_G2L_B128(urow,     &utile[row][seg * 8]);
      ASYNC_G2L_B128(urow + 4, &utile[row][seg * 8 + 4]);
      __builtin_amdgcn_s_wait_asynccnt(0);
    }
#else
#pragma unroll
    for (int j = 0; j < 8; ++j) utile[row][seg * 8 + j] = urow[j];
#endif
    if (t + 1 < Tt) __builtin_prefetch(urow + Dd, 0, 0);  // global_prefetch_b8
    __syncthreads();

    // 2. partial P = U[:,Kw] x B_discT[Kw,:], K=16 per wave -> 4 WMMAs
    {
      v8f c = {};
#pragma unroll
      for (int kk = 0; kk < 4; ++kk) {
        int kb = wv * 16 + kk * 4 + kh * 2;
        v2f a;
        a.x = utile[m][kb];
        a.y = utile[m][kb + 1];
        c = WMMA_F32(a, bf[kk], c);
      }
#pragma unroll
      for (int i = 0; i < 8; ++i) pred[wv][i + kh * 8][m] = c[i];
    }
    __syncthreads();

    // 3. h = clip(h*A_disc + P)
    {
      float p = 0.f;
#pragma unroll
      for (int w = 0; w < 8; ++w) p += pred[w][row][seg];
      float hn = htile[row][seg] * adisc[seg] + p;
      hn = fminf(fmaxf(hn, -10.f), 10.f);
      htile[row][seg] = hn;
    }
    __syncthreads();

    // 4. Y = H x CT (wave's 16-wide d chunk), + u*Dvec, clip
    {
      v8f c = {};
#pragma unroll
      for (int kk = 0; kk < 4; ++kk) {
        int kb = kk * 4 + kh * 2;
        v2f a;
        a.x = htile[m][kb];
        a.y = htile[m][kb + 1];
        c = WMMA_F32(a, cf[kk], c);
      }
#pragma unroll
      for (int i = 0; i < 8; ++i) {
        int r   = i + kh * 8;
        int col = wv * 16 + m;
        float yv = c[i] + utile[r][col] * dv[col];
        yv = fminf(fmaxf(yv, -10.f), 10.f);
        ytile[r][col] = yv;
      }
    }
    __syncthreads();

    // 5. LN1 stats over D
    {
      float s1 = 0.f, s2 = 0.f;
#pragma unroll
      for (int j = 0; j < 8; ++j) {
        float v = ytile[row][seg * 8 + j];
        s1 += v; s2 += v * v;
      }
      rstat[row][seg][0] = s1; rstat[row][seg][1] = s2;
    }
    __syncthreads();
    if (tid < 16) {
      float s1 = 0.f, s2 = 0.f;
      for (int j = 0; j < 16; ++j) { s1 += rstat[tid][j][0]; s2 += rstat[tid][j][1]; }
      float mean = s1 * (1.f / 128.f);
      float var  = s2 * (1.f / 128.f) - mean * mean;
      mstat[tid][0] = mean; mstat[tid][1] = rsqrtf(var + EPS);
    }
    __syncthreads();

    // 6. z = relu(u + LN1(y)); LN2 stats   (u read back from LDS)
    {
      float mean = mstat[row][0], rstd = mstat[row][1];
      float s1 = 0.f, s2 = 0.f;
#pragma unroll
      for (int j = 0; j < 8; ++j) {
        int d = seg * 8 + j;
        float s4 = (ytile[row][d] - mean) * rstd * sg1[d] + sb1[d];
        float z  = fmaxf(utile[row][d] + s4, 0.f);
        ytile[row][d] = z;
        s1 += z; s2 += z * z;
      }
      rstat[row][seg][0] = s1; rstat[row][seg][1] = s2;
    }
    __syncthreads();
    if (tid < 16) {
      float s1 = 0.f, s2 = 0.f;
      for (int j = 0; j < 16; ++j) { s1 += rstat[tid][j][0]; s2 += rstat[tid][j][1]; }
      float mean = s1 * (1.f / 128.f);
      float var  = s2 * (1.f / 128.f) - mean * mean;
      mstat[tid][0] = mean; mstat[tid][1] = rsqrtf(var + EPS);
    }
    __syncthreads();

    // 7. pooled accumulation of LN2(z)
    {
      float mean = mstat[row][0], rstd = mstat[row][1];
#pragma unroll
      for (int j = 0; j < 8; ++j) {
        int d = seg * 8 + j;
        acc[j] += (ytile[row][d] - mean) * rstd * sg2[d] + sb2[d];
      }
    }
  }

#pragma unroll
  for (int j = 0; j < 8; ++j) {
    int d = seg * 8 + j;
    pooled[(size_t)(b0 + row) * Dd + d] = acc[j] * (1.f / (float)Tt);
  }
}

// ---------------------------------------------------------------------------
// Kernel 3: MLP head (tiny: 128x128 @ 128x64 -> relu -> @ 64x3)
// ---------------------------------------------------------------------------
__global__ __launch_bounds__(128)
void s4_head(const float* __restrict__ pooled, const float* __restrict__ W1,
             const float* __restrict__ b1, const float* __restrict__ W2,
             const float* __restrict__ b2, float* __restrict__ out) {
  int b = threadIdx.x;
  float hid[64];
  const float* pr = pooled + (size_t)b * Dd;
  for (int j = 0; j < 64; ++j) hid[j] = b1[j];
  for (int d = 0; d < 128; ++d) {
    float pv = pr[d];
    const float* wrow = W1 + (size_t)d * 64;
    for (int j = 0; j < 64; ++j) hid[j] += pv * wrow[j];
  }
  for (int j = 0; j < 64; ++j) hid[j] = fmaxf(hid[j], 0.f);
  for (int c = 0; c < 3; ++c) {
    float o = b2[c];
    for (int j = 0; j < 64; ++j) o += hid[j] * W2[j * 3 + c];
    out[b * 3 + c] = o;
  }
}

extern "C" void kernel_launch(void* const* d_in, const int* in_sizes, int n_in,
                              void* d_out, int out_size, void* d_ws, size_t ws_size,
                              hipStream_t stream) {
  const float* x     = (const float*)d_in[0];
  const float* convw = (const float*)d_in[1];
  const float* convb = (const float*)d_in[2];
  const float* bng   = (const float*)d_in[3];
  const float* bnb   = (const float*)d_in[4];
  const float* Win   = (const float*)d_in[5];
  const float* bin   = (const float*)d_in[6];
  const float* pos   = (const float*)d_in[7];
  const float* Aarr  = (const float*)d_in[8];
  const float* Bmat  = (const float*)d_in[9];
  const float* Cmat  = (const float*)d_in[10];
  const float* Dvec  = (const float*)d_in[11];
  const float* g1    = (const float*)d_in[12];
  const float* be1   = (const float*)d_in[13];
  const float* g2    = (const float*)d_in[14];
  const float* be2   = (const float*)d_in[15];
  const float* W1    = (const float*)d_in[16];
  const float* b1    = (const float*)d_in[17];
  const float* W2    = (const float*)d_in[18];
  const float* b2    = (const float*)d_in[19];

  float* hin    = (float*)d_ws;                       // [128, 2000, 128] f32
  float* pooled = hin + (size_t)Bn * Tt * Dd;         // [128, 128] f32

  s4_conv_proj<<<dim3(Bn * 32), dim3(128), 0, stream>>>(x, convw, convb, bng, bnb,
                                                        Win, bin, pos, hin);
  s4_scan<<<dim3(Bn / 16), dim3(256), 0, stream>>>(hin, Aarr, Bmat, Cmat, Dvec,
                                                   g1, be1, g2, be2, pooled);
  s4_head<<<dim3(1), dim3(128), 0, stream>>>(pooled, W1, b1, W2, b2, (float*)d_out);
}